// FOFE_Encoding_19000935317529
// MI455X (gfx1250) — compile-verified
//
#include <hip/hip_runtime.h>

// FOFE encoding: B=512 sentences, S=256 words, W=24 chars/word, V=256 vocab.
// z[word, v] = sum_k (chars[word,k]!=0) * alpha^(#non-pad after k) * [chars[word,k]==v]
// Output tuple: (z [B*S*V f32], lengths [B i32] appended bitwise).

#define FOFE_B 512
#define FOFE_S 256
#define FOFE_W 24
#define FOFE_V 256
#define WPB    8            // one wave32 per word, 8 words per 256-thread block

__global__ __launch_bounds__(32 * WPB)
void fofe_kernel(const int* __restrict__ chars,
                 const float* __restrict__ alpha_p,
                 float* __restrict__ z)
{
    __shared__ float hist[WPB][FOFE_V];     // 8 KB: one 256-bin histogram per wave

    const int lane = threadIdx.x & 31;
    const int wv   = threadIdx.x >> 5;
    const int word = blockIdx.x * WPB + wv; // word index in [0, B*S)

    // Zero this wave's private histogram (stride-1 across lanes: conflict-free).
    #pragma unroll
    for (int j = 0; j < 8; ++j)
        hist[wv][lane + 32 * j] = 0.0f;

    // Lanes 0..23 each own one char slot of this word (coalesced 96B/wave load).
    int c = 0;
    if (lane < FOFE_W)
        c = chars[word * FOFE_W + lane];

    // Wave-wide non-pad mask via wave32 ballot; suffix count via popcount.
#if __has_builtin(__builtin_amdgcn_ballot_w32)
    unsigned nz = __builtin_amdgcn_ballot_w32(c != 0) & 0x00FFFFFFu;
#else
    unsigned nz = (unsigned)__ballot(c != 0) & 0x00FFFFFFu;
#endif

    float wgt = 0.0f;
    if (lane < FOFE_W && c != 0) {
        int   n_after = __builtin_popcount(nz >> (lane + 1)); // strictly-after count
        float log2a   = log2f(alpha_p[0]);
        wgt = exp2f((float)n_after * log2a);                  // alpha^n_after
    }

    // Scatter-add into per-word LDS histogram (ds_add_f32). Weights within a
    // word are distinct powers of alpha -> exact, order-independent f32 sum.
    if (lane < FOFE_W)
        atomicAdd(&hist[wv][c & (FOFE_V - 1)], wgt);

    // Ensure LDS atomics have landed before the async LDS->global engine reads.
#if __has_builtin(__builtin_amdgcn_s_wait_dscnt)
    __builtin_amdgcn_s_wait_dscnt(0);
#else
    asm volatile("s_wait_dscnt 0" ::: "memory");
#endif

    // Each lane owns 8 consecutive vocab bins = 32B; wave writes 1KB contiguous.
    float*       gdst = z + (size_t)word * FOFE_V + lane * 8;
    const float* lsrc = &hist[wv][lane * 8];

#if __has_builtin(__builtin_amdgcn_global_store_async_from_lds_b128)
    // Builtin signature (from hipcc diagnostics): pointers to int4-vector in
    // the global (AS1) and LDS (AS3) address spaces, + imm offset, imm cpol.
    typedef int v4i __attribute__((vector_size(16)));
    typedef __attribute__((address_space(1))) v4i GV4;
    typedef __attribute__((address_space(3))) v4i LV4;
    // INST_OFFSET applies to both LDS and global addresses (ISA 08 §4.4).
    __builtin_amdgcn_global_store_async_from_lds_b128((GV4*)gdst, (LV4*)lsrc,  0, 0);
    __builtin_amdgcn_global_store_async_from_lds_b128((GV4*)gdst, (LV4*)lsrc, 16, 0);
  #if __has_builtin(__builtin_amdgcn_s_wait_asynccnt)
    __builtin_amdgcn_s_wait_asynccnt(0);
  #else
    asm volatile("s_wait_asynccnt 0" ::: "memory");
  #endif
#else
    // Fallback: VGPR bounce (ds_load_b128 + global_store_b128 x2).
    float4 v0 = *reinterpret_cast<const float4*>(lsrc);
    float4 v1 = *reinterpret_cast<const float4*>(lsrc + 4);
    *reinterpret_cast<float4*>(gdst)     = v0;
    *reinterpret_cast<float4*>(gdst + 4) = v1;
#endif
}

extern "C" void kernel_launch(void* const* d_in, const int* in_sizes, int n_in,
                              void* d_out, int out_size, void* d_ws, size_t ws_size,
                              hipStream_t stream) {
    const int*   chars   = (const int*)d_in[0];
    const int*   lengths = (const int*)d_in[1];
    const float* alpha   = (const float*)d_in[2];
    float*       z       = (float*)d_out;

    const int words = FOFE_B * FOFE_S;          // 131072 words, one wave each
    dim3 grid(words / WPB);                      // 16384 blocks
    dim3 block(32 * WPB);                        // 256 threads = 8 wave32

    hipLaunchKernelGGL(fofe_kernel, grid, block, 0, stream, chars, alpha, z);

    // Second tuple output: lengths passthrough, appended bitwise after z.
    (void)hipMemcpyAsync(z + (size_t)FOFE_B * FOFE_S * FOFE_V, lengths,
                         (size_t)in_sizes[1] * sizeof(int),
                         hipMemcpyDeviceToDevice, stream);
}